// Head_77584289235603
// MI455X (gfx1250) — compile-verified
//
#include <hip/hip_runtime.h>

typedef unsigned long long ull;
typedef __attribute__((ext_vector_type(2))) float v2f;
typedef __attribute__((ext_vector_type(8))) float v8f;

#define NBI   16          // batch
#define NCLS  80
#define NBOX  25200       // per image
#define NSC   (NBOX*NCLS) // 2,016,000 per image
#define NCAND 512
#define NDET  100
#define NBINS 1024
#define BCAP  4096
#define SCORE_T 0.25f
#define NMS_T   0.45f
#define MAXSZ   640.0f

// ---- workspace layout (bytes) ----
#define OFF_HIST   0u                          // NBI*NBINS int   = 65536
#define OFF_META   (OFF_HIST + NBI*NBINS*4u)   // NBI*4 int       = 256
#define OFF_SURE   (OFF_META + NBI*4u*4u)      // NBI*512 ull     = 65536
#define OFF_BORD   (OFF_SURE + NBI*NCAND*8u)   // NBI*4096 ull    = 524288
#define OFF_CSC    (OFF_BORD + NBI*BCAP*8u)    // NBI*512 float
#define OFF_CIDX   (OFF_CSC  + NBI*NCAND*4u)   // NBI*512 int

__device__ __forceinline__ float sigf(float x) { return 1.0f / (1.0f + __expf(-x)); }

__device__ __forceinline__ const float* elem_ptr(const float* p0, const float* p1,
                                                 const float* p2, int b, int n,
                                                 int& lvl, int& a, int& gx, int& gy,
                                                 float& stride) {
  const float* p; int H, W;
  if (n < 19200)      { lvl = 0; p = p0; H = 80; W = 80; stride = 8.f;  }
  else if (n < 24000) { lvl = 1; p = p1; H = 40; W = 40; stride = 16.f; n -= 19200; }
  else                { lvl = 2; p = p2; H = 20; W = 20; stride = 32.f; n -= 24000; }
  a = n % 3; int cell = n / 3; gx = cell % W; gy = cell / W;
  return p + ((((size_t)b * H + gy) * W + gx) * 3 + a) * 85;
}

__device__ __forceinline__ float score_at(const float* p0, const float* p1,
                                          const float* p2, int b, int f) {
  int n = f / NCLS, c = f - n * NCLS;
  int lvl, a, gx, gy; float st;
  const float* e = elem_ptr(p0, p1, p2, b, n, lvl, a, gx, gy, st);
  return sigf(e[4]) * sigf(e[5 + c]);
}

__device__ __forceinline__ void decode_box(const float* p0, const float* p1,
                                           const float* p2, const float* anch,
                                           int b, int n, float imw, float imh,
                                           float out[4]) {
  int lvl, a, gx, gy; float st;
  const float* e = elem_ptr(p0, p1, p2, b, n, lvl, a, gx, gy, st);
  float sx = sigf(e[0]), sy = sigf(e[1]), sw = sigf(e[2]), sh = sigf(e[3]);
  float cx = (2.f * sx - 0.5f + (float)gx) * st;
  float cy = (2.f * sy - 0.5f + (float)gy) * st;
  float aw = anch[(lvl * 3 + a) * 2 + 0];
  float ah = anch[(lvl * 3 + a) * 2 + 1];
  float bw = 4.f * sw * sw * aw, bh = 4.f * sh * sh * ah;
  out[0] = fminf(fmaxf(cx - bw * 0.5f, 0.f), imw);
  out[1] = fminf(fmaxf(cy - bh * 0.5f, 0.f), imh);
  out[2] = fminf(fmaxf(cx + bw * 0.5f, 0.f), imw);
  out[3] = fminf(fmaxf(cy + bh * 0.5f, 0.f), imh);
}

__device__ __forceinline__ float iou4(float ax1, float ay1, float ax2, float ay2,
                                      float bx1, float by1, float bx2, float by2) {
  float lx = fmaxf(ax1, bx1), ly = fmaxf(ay1, by1);
  float rx = fminf(ax2, bx2), ry = fminf(ay2, by2);
  float iw = fmaxf(rx - lx, 0.f), ih = fmaxf(ry - ly, 0.f);
  float inter = iw * ih;
  float aa = (ax2 - ax1) * (ay2 - ay1);
  float ab = (bx2 - bx1) * (by2 - by1);
  return inter / (aa + ab - inter + 1e-9f);
}

// ---------- pass 1: per-image score histogram ----------
__global__ void k_hist(const float* p0, const float* p1, const float* p2, int* hist) {
  __shared__ int lh[NBINS];
  int tid = threadIdx.x, b = blockIdx.y;
  for (int i = tid; i < NBINS; i += 256) lh[i] = 0;
  __syncthreads();
  int f = blockIdx.x * 256 + tid;
  float s = score_at(p0, p1, p2, b, f);
  if (s > SCORE_T) {
    int bin = min(NBINS - 1, (int)((s - SCORE_T) * ((float)NBINS / 0.75f)));
    atomicAdd(&lh[bin], 1);
  }
  __syncthreads();
  for (int i = tid; i < NBINS; i += 256)
    if (lh[i]) atomicAdd(&hist[b * NBINS + i], lh[i]);
}

// ---------- pass 2: find cutoff bin per image ----------
__global__ void k_scan(const int* hist, int* meta) {
  int b = blockIdx.x;
  if (threadIdx.x != 0) return;
  const int* h = hist + b * NBINS;
  int cum = 0, kstar = -1, above = 0;
  for (int i = NBINS - 1; i >= 0; --i) {
    int c = h[i];
    if (cum + c >= NCAND) { kstar = i; above = cum; break; }
    cum += c;
  }
  meta[b * 4 + 0] = kstar;
  meta[b * 4 + 1] = above;
  // meta[b*4+2] (sure count) and [b*4+3] (border count) already zeroed by memset
}

// ---------- pass 3: compact sure + border candidates ----------
__global__ void k_compact(const float* p0, const float* p1, const float* p2,
                          int* meta, ull* sure, ull* border) {
  int tid = threadIdx.x, b = blockIdx.y;
  int f = blockIdx.x * 256 + tid;
  float s = score_at(p0, p1, p2, b, f);
  if (s <= SCORE_T) return;
  int bin = min(NBINS - 1, (int)((s - SCORE_T) * ((float)NBINS / 0.75f)));
  int kstar = meta[b * 4 + 0];
  ull key = ((ull)__float_as_uint(s) << 32) | (unsigned)(~f);
  if (bin > kstar) {
    int p = atomicAdd(&meta[b * 4 + 2], 1);
    if (p < NCAND) sure[b * NCAND + p] = key;
  } else if (bin == kstar) {
    int p = atomicAdd(&meta[b * 4 + 3], 1);
    if (p < BCAP) border[b * BCAP + p] = key;
  }
}

// ---------- pass 4: exact top-512 via 4096-wide bitonic sort ----------
__global__ void k_sortsel(const ull* sure, const ull* border, const int* meta,
                          float* csc, int* cidx) {
  __shared__ ull sk[BCAP];
  int b = blockIdx.x, tid = threadIdx.x;
  int ns = min(meta[b * 4 + 2], NCAND);
  int nb = min(meta[b * 4 + 3], BCAP - ns);
  for (int i = tid; i < BCAP; i += 512) {
    ull v = 0;
    if (i < ns)            v = sure[b * NCAND + i];
    else if (i < ns + nb)  v = border[b * BCAP + (i - ns)];
    sk[i] = v;
  }
  __syncthreads();
  for (int k = 2; k <= BCAP; k <<= 1)
    for (int j = k >> 1; j > 0; j >>= 1) {
      for (int t = 0; t < BCAP / 512; ++t) {
        int i = tid + t * 512, ixj = i ^ j;
        if (ixj > i) {
          bool up = ((i & k) == 0);
          ull a = sk[i], c = sk[ixj];
          if (up ? (a < c) : (a > c)) { sk[i] = c; sk[ixj] = a; }
        }
      }
      __syncthreads();
    }
  ull kk = sk[tid];
  if (tid < NCAND) {
    float sc = __uint_as_float((unsigned)(kk >> 32));
    int idx = (int)(~(unsigned)kk);
    if (kk == 0 || idx < 0 || idx >= NSC) { idx = 0; sc = 0.f; }
    csc[b * NCAND + tid] = sc;
    cidx[b * NCAND + tid] = idx;
  }
}

// ---------- pass 5: NMS + top-100 + WMMA box fusion ----------
__global__ void __launch_bounds__(256)
k_finish(const float* p0, const float* p1, const float* p2, const float* anch,
         const int* shapes, const float* csc, const int* cidx, float* out) {
  __shared__ float sbox[NCAND][4];
  __shared__ float scsc[NCAND];
  __shared__ int   slab[NCAND];
  __shared__ int   skeep[NCAND];
  __shared__ ull   skey[NCAND];
  __shared__ float sdbox[NDET][4];
  __shared__ float sdsc[NDET];
  __shared__ int   sdlab[NDET];
  __shared__ float sfus[NDET][5];

  int b = blockIdx.x, tid = threadIdx.x;
  float imh = (float)shapes[b * 2 + 0];
  float imw = (float)shapes[b * 2 + 1];

  // phase 1: decode candidate boxes into LDS
  for (int i = tid; i < NCAND; i += 256) {
    float sc = csc[b * NCAND + i];
    int ci = cidx[b * NCAND + i];
    int lab = ci % NCLS, bx = ci / NCLS;
    float bb[4];
    decode_box(p0, p1, p2, anch, b, bx, imw, imh, bb);
    sbox[i][0] = bb[0]; sbox[i][1] = bb[1]; sbox[i][2] = bb[2]; sbox[i][3] = bb[3];
    scsc[i] = sc; slab[i] = lab; skeep[i] = (sc > 0.f) ? 1 : 0;
  }
  __syncthreads();

  // phase 2: sequential class-offset NMS (reference semantics)
  for (int i = 0; i < NCAND; ++i) {
    __syncthreads();
    if (!skeep[i]) continue;
    float offi = (float)slab[i] * MAXSZ;
    float ax1 = sbox[i][0] + offi, ay1 = sbox[i][1] + offi;
    float ax2 = sbox[i][2] + offi, ay2 = sbox[i][3] + offi;
    for (int j = tid; j < NCAND; j += 256) {
      if (j > i && skeep[j]) {
        float offj = (float)slab[j] * MAXSZ;
        float v = iou4(ax1, ay1, ax2, ay2,
                       sbox[j][0] + offj, sbox[j][1] + offj,
                       sbox[j][2] + offj, sbox[j][3] + offj);
        if (v > NMS_T) skeep[j] = 0;
      }
    }
  }
  __syncthreads();

  // phase 3: top-100 of kept scores (bitonic-512, ties -> lowest index)
  for (int i = tid; i < NCAND; i += 256) {
    float ks = skeep[i] ? scsc[i] : 0.f;
    skey[i] = ((ull)__float_as_uint(ks) << 32) | (unsigned)(~i);
  }
  __syncthreads();
  for (int k = 2; k <= NCAND; k <<= 1)
    for (int j = k >> 1; j > 0; j >>= 1) {
      for (int t = 0; t < 2; ++t) {
        int i = tid + t * 256, ixj = i ^ j;
        if (ixj > i) {
          bool up = ((i & k) == 0);
          ull a = skey[i], c = skey[ixj];
          if (up ? (a < c) : (a > c)) { skey[i] = c; skey[ixj] = a; }
        }
      }
      __syncthreads();
    }
  for (int d = tid; d < NDET; d += 256) {
    ull kk = skey[d];
    float ds = __uint_as_float((unsigned)(kk >> 32));
    int pos = ((int)(~(unsigned)kk)) & (NCAND - 1);
    sdsc[d] = ds; sdlab[d] = slab[pos];
    sdbox[d][0] = sbox[pos][0]; sdbox[d][1] = sbox[pos][1];
    sdbox[d][2] = sbox[pos][2]; sdbox[d][3] = sbox[pos][3];
  }
  __syncthreads();

  // phase 4: fusion GEMM  D(112x16) = W(112x512) * [box|1](512x16pad)
  // via V_WMMA_F32_16X16X4_F32; 7 waves, one 16x16 tile each, 128 k-steps.
  // Fully branchless inner loop: all LDS loads unconditional (clamped column
  // index for B), selections via v_cndmask, no EXEC changes around the WMMA.
  int wave = tid >> 5, lane = tid & 31;
  if (wave < 7) {
    int half = lane >> 4, lm = lane & 15;
    int m = wave * 16 + lm;
    int mc = min(m, NDET - 1);
    float mvf = (m < NDET) ? 1.f : 0.f;
    float dx1 = sdbox[mc][0], dy1 = sdbox[mc][1];
    float dx2 = sdbox[mc][2], dy2 = sdbox[mc][3];
    int lmc = lm & 3;                    // clamped col for unconditional B load
    float btail = (lm == 4) ? 1.f : 0.f; // ones column for denominator
    v8f acc = {};
    for (int s = 0; s < NCAND / 4; ++s) {
      int ka = s * 4 + half * 2; // A: lanes<16 hold K={0,1}, lanes>=16 K={2,3}
      float i0 = iou4(dx1, dy1, dx2, dy2, sbox[ka][0], sbox[ka][1], sbox[ka][2], sbox[ka][3]);
      float i1 = iou4(dx1, dy1, dx2, dy2, sbox[ka+1][0], sbox[ka+1][1], sbox[ka+1][2], sbox[ka+1][3]);
      float w0 = scsc[ka] * mvf;     // unconditional loads, cndmask select
      float w1 = scsc[ka + 1] * mvf;
      float a0 = (i0 > NMS_T) ? w0 : 0.f;
      float a1 = (i1 > NMS_T) ? w1 : 0.f;
      // B: vgpr v, lanes<16 row K=v, lanes>=16 row K=v+2; col = lane%16
      int kb = s * 4 + half * 2;
      float v0 = sbox[kb][lmc];        // every lane loads (valid address)
      float v1 = sbox[kb + 1][lmc];
      float b0 = (lm < 4) ? v0 : btail;
      float b1 = (lm < 4) ? v1 : btail;
      v2f A = {a0, a1};
      v2f Bv = {b0, b1};
      acc = __builtin_amdgcn_wmma_f32_16x16x4_f32(false, A, false, Bv,
                                                  (short)0, acc, false, false);
    }
    // D layout: vgpr v -> row v + 8*half, col = lane%16
    for (int v = 0; v < 8; ++v) {
      int r = wave * 16 + v + half * 8;
      if (r < NDET && lm < 5) sfus[r][lm] = acc[v];
    }
  }
  __syncthreads();

  // phase 5: normalize + write outputs
  for (int d = tid; d < NDET; d += 256) {
    float den = sfus[d][4] + 1e-9f;
    float msk = (sdsc[d] > 0.f) ? 1.f : 0.f;
    float* o = out + ((size_t)b * NDET + d) * 5;
    o[0] = sfus[d][0] / den * msk;
    o[1] = sfus[d][1] / den * msk;
    o[2] = sfus[d][2] / den * msk;
    o[3] = sfus[d][3] / den * msk;
    o[4] = sdsc[d];
    out[(size_t)NBI * NDET * 5 + b * NDET + d] = (float)sdlab[d];
  }
}

extern "C" void kernel_launch(void* const* d_in, const int* in_sizes, int n_in,
                              void* d_out, int out_size, void* d_ws, size_t ws_size,
                              hipStream_t stream) {
  const float* p0     = (const float*)d_in[0];
  const float* p1     = (const float*)d_in[1];
  const float* p2     = (const float*)d_in[2];
  const float* anch   = (const float*)d_in[3];
  const int*   shapes = (const int*)d_in[4];
  float* out = (float*)d_out;
  char* ws = (char*)d_ws;
  int*  hist   = (int*)(ws + OFF_HIST);
  int*  meta   = (int*)(ws + OFF_META);
  ull*  sure   = (ull*)(ws + OFF_SURE);
  ull*  border = (ull*)(ws + OFF_BORD);
  float* csc   = (float*)(ws + OFF_CSC);
  int*   cidx  = (int*)(ws + OFF_CIDX);

  // zero histograms + meta counters (capture-safe)
  hipMemsetAsync(ws, 0, OFF_SURE, stream);

  dim3 g1(NSC / 256, NBI);
  k_hist<<<g1, 256, 0, stream>>>(p0, p1, p2, hist);
  k_scan<<<NBI, 32, 0, stream>>>(hist, meta);
  k_compact<<<g1, 256, 0, stream>>>(p0, p1, p2, meta, sure, border);
  k_sortsel<<<NBI, 512, 0, stream>>>(sure, border, meta, csc, cidx);
  k_finish<<<NBI, 256, 0, stream>>>(p0, p1, p2, anch, shapes, csc, cidx, out);
}